// MultiboxLoss_19061064860371
// MI455X (gfx1250) — compile-verified
//
#include <hip/hip_runtime.h>
#include <hip/hip_bf16.h>
#include <float.h>
#include <stdint.h>

// ---------------- problem constants (match reference setup_inputs) ----------
#define B_N   128
#define P_N   8732
#define C_N   21
#define K1_BLK 256
#define K1_GX  ((P_N + K1_BLK - 1) / K1_BLK)   // 35
#define NBLK   (K1_GX * B_N)                   // 4480
#define K2_BLK 1024

// ---------------- CDNA5 async global->LDS path ------------------------------
// gfx1250 ASYNCcnt path: stage streaming data straight into LDS without a
// VGPR round trip. Builtin signatures (probe-verified via clang diagnostics):
//   b32 : (int  addrspace(1)*, int  addrspace(3)*, imm offset, imm cpol)
//   b128: (v4i  addrspace(1)*, v4i  addrspace(3)*, imm offset, imm cpol)
#if defined(__has_builtin)
#if __has_builtin(__builtin_amdgcn_s_wait_asynccnt)
#if __has_builtin(__builtin_amdgcn_global_load_async_to_lds_b128)
#define USE_ASYNC_B128 1
#elif __has_builtin(__builtin_amdgcn_global_load_async_to_lds_b32)
#define USE_ASYNC_B32 1
#endif
#endif
#endif

typedef int v4i __attribute__((vector_size(16)));
typedef __attribute__((address_space(1))) int* as1_int;
typedef __attribute__((address_space(3))) int* as3_int;
typedef __attribute__((address_space(1))) v4i* as1_v4i;
typedef __attribute__((address_space(3))) v4i* as3_v4i;

// Copy 16 bytes (one float4) global -> LDS. Both addresses must be 16B-aligned.
__device__ __forceinline__ void lds_copy_b128(void* lds_dst, const void* g_src) {
#if defined(USE_ASYNC_B128)
  __builtin_amdgcn_global_load_async_to_lds_b128(
      (as1_v4i)(uintptr_t)g_src, (as3_v4i)(uintptr_t)lds_dst, 0, 0);
#elif defined(USE_ASYNC_B32)
#pragma unroll
  for (int q = 0; q < 4; ++q)
    __builtin_amdgcn_global_load_async_to_lds_b32(
        (as1_int)((uintptr_t)g_src + 4u * q),
        (as3_int)((uintptr_t)lds_dst + 4u * q), 0, 0);
#else
  *(uint4*)lds_dst = *(const uint4*)g_src;
#endif
}

__device__ __forceinline__ void lds_copy_wait() {
#if defined(USE_ASYNC_B128) || defined(USE_ASYNC_B32)
  __builtin_amdgcn_s_wait_asynccnt(0);   // s_wait_asynccnt 0
#endif
  __syncthreads();
}

// ---------------- smooth-ex loss --------------------------------------------
// a = 1/(1-e^-2), c = a - 1 - 0.5
__device__ __forceinline__ float smooth_ex(float d) {
  d = fabsf(d);
  const float A = 1.1565176427496657f;
  const float C = -0.3434823572503343f;
  return (d < 1.0f) ? A * (d + (__expf(-2.0f * d) - 1.0f) * 0.5f) : (d + C);
}

// ---------------- kernel 0: zero the atomically-updated counters ------------
__global__ void k_init(int* __restrict__ num_pos_row) {
  const int t = blockIdx.x * blockDim.x + threadIdx.x;
  if (t < B_N) num_pos_row[t] = 0;
}

// ---------------- kernel 1: per-prior logsoftmax / CE / smooth loss ---------
// One block = 256 priors of one batch row. Confidence tile (256x21 f32,
// 21 KB) staged into LDS with async b128 copies (tile base is provably
// 16B-aligned: (b*8732 + 256k)*21 = 0 mod 4 floats; tile sizes 5376/588 are
// multiples of 4). Stride-21 LDS reads are bank-conflict free (gcd(21,64)=1).
__global__ __launch_bounds__(K1_BLK) void k_logsoftmax_pos(
    const float* __restrict__ conf, const float* __restrict__ pred,
    const int* __restrict__ labels, const float* __restrict__ gt,
    float* __restrict__ bg, int* __restrict__ num_pos_row,
    float* __restrict__ ce_part, float* __restrict__ sm_part)
{
  __shared__ __align__(16) float s_conf[K1_BLK * C_N];
  __shared__ float s_rf[2 * (K1_BLK / 32)];
  __shared__ int   s_ri[K1_BLK / 32];

  const int b   = blockIdx.y;
  const int p0  = blockIdx.x * K1_BLK;
  const int tid = threadIdx.x;
  const int nvalid = min(K1_BLK, P_N - p0);
  const int nvec   = (nvalid * C_N) >> 2;        // float4 chunks (exact)
  const float* gsrc = conf + ((size_t)b * P_N + p0) * C_N;
  for (int i = tid; i < nvec; i += K1_BLK)
    lds_copy_b128(&s_conf[i * 4], &gsrc[i * 4]);
  lds_copy_wait();

  float ce = 0.f, sm = 0.f; int np = 0;
  if (tid < nvalid) {
    const size_t gp = (size_t)b * P_N + p0 + tid;
    const float* x = &s_conf[tid * C_N];
    float m = x[0];
#pragma unroll
    for (int c = 1; c < C_N; ++c) m = fmaxf(m, x[c]);
    float se = 0.f;
#pragma unroll
    for (int c = 0; c < C_N; ++c) se += __expf(x[c] - m);
    const float lse = m + __logf(se);           // logsumexp
    const int  lab = labels[gp];
    const bool pos = lab > 0;
    // background loss for mining; positives -> -FLT_MAX sentinel
    bg[gp] = pos ? -FLT_MAX : (lse - x[0]);
    if (pos) {
      np = 1;
      ce = lse - x[lab];                        // -log p(label)
      const float4 pr = *(const float4*)(pred + gp * 4);
      const float4 g4 = *(const float4*)(gt   + gp * 4);
      sm = smooth_ex(pr.x - g4.x) + smooth_ex(pr.y - g4.y)
         + smooth_ex(pr.z - g4.z) + smooth_ex(pr.w - g4.w);
    }
  }
  // wave32 tree reduce, then cross-wave via LDS
#pragma unroll
  for (int off = 16; off > 0; off >>= 1) {
    ce += __shfl_down(ce, off, 32);
    sm += __shfl_down(sm, off, 32);
    np += __shfl_down(np, off, 32);
  }
  const int wave = tid >> 5, lane = tid & 31, nw = K1_BLK / 32;
  if (lane == 0) { s_rf[wave] = ce; s_rf[nw + wave] = sm; s_ri[wave] = np; }
  __syncthreads();
  if (tid == 0) {
    float tce = 0.f, tsm = 0.f; int tnp = 0;
    for (int w = 0; w < nw; ++w) { tce += s_rf[w]; tsm += s_rf[nw + w]; tnp += s_ri[w]; }
    const int blk = b * gridDim.x + blockIdx.x;
    ce_part[blk] = tce;                 // deterministic per-block partials
    sm_part[blk] = tsm;
    if (tnp) atomicAdd(&num_pos_row[b], tnp);   // integer atomic: deterministic
  }
}

// ---------------- kernel 2: hard-negative mining (radix top-k sum) ----------
// One block per batch row; row (8732 f32 = exactly 2183 float4, 16B-aligned)
// staged via async b128 copies. Keys are order-preserving uint transforms of
// the bg losses (positives -> key 0). 32-step bitwise select finds the K-th
// largest key; exact tie handling via (K - count_gt) * value(t).
__device__ __forceinline__ float key_to_float(unsigned k) {
  const unsigned fb = (k & 0x80000000u) ? (k & 0x7FFFFFFFu) : ~k;
  return __uint_as_float(fb);
}

__global__ __launch_bounds__(K2_BLK) void k_mine_topk(
    const float* __restrict__ bg, const int* __restrict__ num_pos_row,
    float* __restrict__ neg_row)
{
  __shared__ __align__(16) unsigned s_key[P_N];   // ~34.9 KB
  __shared__ float s_rf[K2_BLK / 32];
  __shared__ int   s_ri[K2_BLK / 32];

  const int b   = blockIdx.x;
  const int tid = threadIdx.x;
  const float* row = bg + (size_t)b * P_N;
  for (int i = tid; i < (P_N >> 2); i += K2_BLK)
    lds_copy_b128(&s_key[i * 4], &row[i * 4]);
  lds_copy_wait();

  // float bits -> order-preserving key; -FLT_MAX (positive prior) -> 0
  for (int i = tid; i < P_N; i += K2_BLK) {
    const unsigned fb = s_key[i];
    unsigned k = (fb & 0x80000000u) ? ~fb : (fb | 0x80000000u);
    if (fb == 0xFF7FFFFFu) k = 0u;
    s_key[i] = k;
  }
  __syncthreads();

  auto count_ge = [&](unsigned thr) -> int {
    int c = 0;
    for (int i = tid; i < P_N; i += K2_BLK) c += (s_key[i] >= thr);
#pragma unroll
    for (int off = 16; off > 0; off >>= 1) c += __shfl_down(c, off, 32);
    __syncthreads();
    if ((tid & 31) == 0) s_ri[tid >> 5] = c;
    __syncthreads();
    int t = 0;
    for (int w = 0; w < K2_BLK / 32; ++w) t += s_ri[w];
    return t;                                   // broadcast to all threads
  };
  auto reduce_f = [&](float v) -> float {
#pragma unroll
    for (int off = 16; off > 0; off >>= 1) v += __shfl_down(v, off, 32);
    __syncthreads();
    if ((tid & 31) == 0) s_rf[tid >> 5] = v;
    __syncthreads();
    float t = 0.f;
    for (int w = 0; w < K2_BLK / 32; ++w) t += s_rf[w];
    return t;
  };

  const int K = min(3 * num_pos_row[b], P_N);
  float neg = 0.f;
  if (K > 0) {
    const int nf = count_ge(1u);                // #finite negatives
    if (K >= nf) {                              // take every negative
      float sum = 0.f;
      for (int i = tid; i < P_N; i += K2_BLK) {
        const unsigned k = s_key[i];
        if (k > 0u) sum += key_to_float(k);
      }
      neg = reduce_f(sum);
    } else {                                    // bitwise K-th largest select
      unsigned t = 0u;
      for (int bit = 31; bit >= 0; --bit) {
        const unsigned cand = t | (1u << bit);
        if (count_ge(cand) >= K) t = cand;
      }
      float sum = 0.f;
      for (int i = tid; i < P_N; i += K2_BLK) {
        const unsigned k = s_key[i];
        if (k > t) sum += key_to_float(k);
      }
      sum = reduce_f(sum);
      const int cgt = count_ge(t + 1u);         // strictly-greater count
      neg = sum + (float)(K - cgt) * key_to_float(t);
    }
  }
  if (tid == 0) neg_row[b] = neg;
}

// ---------------- kernel 3: deterministic final reduction -------------------
__global__ __launch_bounds__(256) void k_final(
    const float* __restrict__ ce_part, const float* __restrict__ sm_part,
    int nblk, const int* __restrict__ num_pos_row,
    const float* __restrict__ neg_row, float* __restrict__ out)
{
  __shared__ float s_ce[256], s_sm[256], s_ng[256];
  __shared__ int   s_np[256];
  const int t = threadIdx.x;
  float ce = 0.f, sm = 0.f, ng = 0.f; int np = 0;
  for (int i = t; i < nblk; i += 256) { ce += ce_part[i]; sm += sm_part[i]; }
  for (int i = t; i < B_N; i += 256)  { np += num_pos_row[i]; ng += neg_row[i]; }
  s_ce[t] = ce; s_sm[t] = sm; s_ng[t] = ng; s_np[t] = np;
  __syncthreads();
  for (int off = 128; off > 0; off >>= 1) {
    if (t < off) {
      s_ce[t] += s_ce[t + off]; s_sm[t] += s_sm[t + off];
      s_ng[t] += s_ng[t + off]; s_np[t] += s_np[t + off];
    }
    __syncthreads();
  }
  if (t == 0) {
    const float npf = (float)s_np[0];
    out[0] = s_sm[0] / npf;                       // smooth_sum / n_pos
    out[1] = (s_ce[0] + s_ng[0]) / npf;           // classification / n_pos
  }
}

// ---------------- host launch ------------------------------------------------
extern "C" void kernel_launch(void* const* d_in, const int* in_sizes, int n_in,
                              void* d_out, int out_size, void* d_ws, size_t ws_size,
                              hipStream_t stream) {
  (void)in_sizes; (void)n_in; (void)out_size; (void)ws_size;
  const float* conf   = (const float*)d_in[0];
  const float* pred   = (const float*)d_in[1];
  const int*   labels = (const int*)d_in[2];
  const float* gt     = (const float*)d_in[3];
  float*       out    = (float*)d_out;

  char* ws = (char*)d_ws;
  size_t off = 0;
  float* bg = (float*)(ws + off);          off += (size_t)B_N * P_N * sizeof(float);
  off = (off + 255) & ~(size_t)255;
  int* num_pos_row = (int*)(ws + off);     off += B_N * sizeof(int);
  off = (off + 255) & ~(size_t)255;
  float* ce_part = (float*)(ws + off);     off += NBLK * sizeof(float);
  off = (off + 255) & ~(size_t)255;
  float* sm_part = (float*)(ws + off);     off += NBLK * sizeof(float);
  off = (off + 255) & ~(size_t)255;
  float* neg_row = (float*)(ws + off);     off += B_N * sizeof(float);

  k_init<<<1, 256, 0, stream>>>(num_pos_row);
  dim3 g1(K1_GX, B_N);
  k_logsoftmax_pos<<<g1, K1_BLK, 0, stream>>>(conf, pred, labels, gt,
                                              bg, num_pos_row, ce_part, sm_part);
  k_mine_topk<<<B_N, K2_BLK, 0, stream>>>(bg, num_pos_row, neg_row);
  k_final<<<1, 256, 0, stream>>>(ce_part, sm_part, NBLK, num_pos_row, neg_row, out);
}